// MultiHeadSelfAttention_1846835937398
// MI455X (gfx1250) — compile-verified
//
#include <hip/hip_runtime.h>
#include <hip/hip_bf16.h>

// MI455X / gfx1250, wave32. fp32 WMMA (V_WMMA_F32_16X16X4_F32) for reference
// precision + async global->LDS double buffering (ASYNCcnt) for data movement.

typedef __attribute__((ext_vector_type(2))) float v2f;
typedef __attribute__((ext_vector_type(8))) float v8f;
typedef __attribute__((__vector_size__(16))) int i32x4;

#define BB 2
#define TT 2048
#define CC 1024
#define HH 16
#define HDIM 64
#define MM (BB * TT)

#define AS1 __attribute__((address_space(1)))
#define AS3 __attribute__((address_space(3)))

#if defined(__has_builtin)
#if __has_builtin(__builtin_amdgcn_global_load_async_to_lds_b128)
#define HAVE_ASYNC_BUILTIN 1
#endif
#if __has_builtin(__builtin_amdgcn_s_wait_asynccnt)
#define HAVE_ASYNCCNT_BUILTIN 1
#endif
#endif

// copy 16 bytes global -> LDS, tracked by ASYNCcnt (no VGPR staging)
__device__ __forceinline__ void async_ld16(const float* g, float* l) {
#ifdef HAVE_ASYNC_BUILTIN
  __builtin_amdgcn_global_load_async_to_lds_b128((AS1 i32x4*)g, (AS3 i32x4*)l,
                                                 0, 0);
#else
  uint32_t loff = (uint32_t)(uintptr_t)(AS3 void*)l;  // LDS byte offset
  asm volatile("global_load_async_to_lds_b128 %0, %1, off" ::"v"(loff),
               "v"((uint64_t)(uintptr_t)g)
               : "memory");
#endif
}

__device__ __forceinline__ void async_wait0() {
#ifdef HAVE_ASYNCCNT_BUILTIN
  __builtin_amdgcn_s_wait_asynccnt(0);
#else
  asm volatile("s_wait_asynccnt 0x0" ::: "memory");
#endif
  asm volatile("" ::: "memory");
}

__device__ __forceinline__ v8f wmma4f32(v2f a, v2f b, v8f c) {
  return __builtin_amdgcn_wmma_f32_16x16x4_f32(false, a, false, b, (short)0, c,
                                               false, false);
}

// ---------------------------------------------------------------------------
// GEMM + bias: Out[M,N] = A[M,K] @ W[K,N] + bias[N]
// Block tile 128x128, K-tile 32, 256 threads = 8 waves (4m x 2n),
// wave tile 32x64 = 2x4 WMMA subtiles. Double-buffered async LDS staging.
// ---------------------------------------------------------------------------
#define LDA 36   // 32 + 4 pad; row stride 144B (16B aligned, conflict-free)
#define LDB 136  // 128 + 8 pad; row stride 544B

__global__ __launch_bounds__(256) void gemm_bias_wmma(
    const float* __restrict__ A, const float* __restrict__ W,
    const float* __restrict__ bias, float* __restrict__ Out, int M, int N,
    int K) {
  __shared__ float As[2][128 * LDA];
  __shared__ float Bs[2][32 * LDB];

  const int tid = threadIdx.x;
  const int lane = tid & 31;
  const int wid = tid >> 5;
  const int half = lane >> 4;
  const int ln = lane & 15;
  const int wm = (wid >> 1) * 32;  // wave M offset (0,32,64,96)
  const int wn = (wid & 1) * 64;   // wave N offset (0,64)

  const int m0 = blockIdx.y * 128;
  const int n0 = blockIdx.x * 128;

  v8f acc[2][4];
#pragma unroll
  for (int i = 0; i < 2; ++i)
#pragma unroll
    for (int j = 0; j < 4; ++j)
      acc[i][j] = (v8f){0.f, 0.f, 0.f, 0.f, 0.f, 0.f, 0.f, 0.f};

  const int nk = K / 32;

  // stage k-tile `kt` into buffer `buf` (async, per-lane 16B chunks)
  auto stage = [&](int buf, int kt) {
#pragma unroll
    for (int i = 0; i < 4; ++i) {  // A tile 128x32
      int c = tid + i * 256;
      int row = c >> 3, col = (c & 7) * 4;
      async_ld16(A + (size_t)(m0 + row) * K + kt + col,
                 &As[buf][row * LDA + col]);
    }
#pragma unroll
    for (int i = 0; i < 4; ++i) {  // W tile 32x128
      int c = tid + i * 256;
      int row = c >> 5, col = (c & 31) * 4;
      async_ld16(W + (size_t)(kt + row) * N + n0 + col,
                 &Bs[buf][row * LDB + col]);
    }
  };

  stage(0, 0);
  async_wait0();
  __syncthreads();

  for (int t = 0; t < nk; ++t) {
    if (t + 1 < nk) stage((t + 1) & 1, (t + 1) * 32);
    const float* Ab = &As[t & 1][0];
    const float* Bb = &Bs[t & 1][0];

#pragma unroll
    for (int s = 0; s < 8; ++s) {
      const int k0 = 4 * s + 2 * half;
      v2f af[2], bf[4];
#pragma unroll
      for (int ms = 0; ms < 2; ++ms)
        af[ms] = *(const v2f*)&Ab[(wm + 16 * ms + ln) * LDA + k0];
#pragma unroll
      for (int ns = 0; ns < 4; ++ns) {
        v2f tb;
        tb.x = Bb[k0 * LDB + wn + 16 * ns + ln];
        tb.y = Bb[(k0 + 1) * LDB + wn + 16 * ns + ln];
        bf[ns] = tb;
      }
#pragma unroll
      for (int ms = 0; ms < 2; ++ms)
#pragma unroll
        for (int ns = 0; ns < 4; ++ns)
          acc[ms][ns] = wmma4f32(af[ms], bf[ns], acc[ms][ns]);
    }
    async_wait0();   // next tile resident before anyone flips buffers
    __syncthreads(); // all waves done reading current buffers
  }

#pragma unroll
  for (int ms = 0; ms < 2; ++ms)
#pragma unroll
    for (int ns = 0; ns < 4; ++ns) {
      const int col = n0 + wn + 16 * ns + ln;
      const float bv = bias[col];
#pragma unroll
      for (int r = 0; r < 8; ++r) {
        int row = m0 + wm + 16 * ms + r + 8 * half;
        Out[(size_t)row * N + col] = acc[ms][ns][r] + bv;
      }
    }
}

// ---------------------------------------------------------------------------
// Causal flash attention. Grid (T/64, H, B), 128 threads = 4 waves.
// Wave w owns query rows [q0+16w, q0+16w+16). Double-buffered async K/V.
// ---------------------------------------------------------------------------
#define LDK 68  // 64 + 4 pad; 272B rows
#define LDV 72  // 64 + 8 pad; 288B rows
#define LDP 68

__global__ __launch_bounds__(128) void attn_wmma(const float* __restrict__ qkv,
                                                 float* __restrict__ ctx) {
  __shared__ float Ks[2][64 * LDK];
  __shared__ float Vs[2][64 * LDV];
  __shared__ float Ps[4][16 * LDP];

  const int tid = threadIdx.x;
  const int lane = tid & 31;
  const int w = tid >> 5;
  const int half = lane >> 4;
  const int ln = lane & 15;

  const int qb = blockIdx.x;
  const int h = blockIdx.y;
  const int b = blockIdx.z;
  const int q0 = qb * 64;

  const float scale = 0.125f;  // HD^-0.5 folded into Q

  // Q A-fragments for this wave's 16 query rows
  const int qrow = q0 + w * 16 + ln;
  const float* qp = qkv + (size_t)(b * TT + qrow) * (3 * CC) + h * HDIM;
  v2f qf[16];
#pragma unroll
  for (int s = 0; s < 16; ++s) {
    int k0 = 4 * s + 2 * half;
    v2f t;
    t.x = qp[k0] * scale;
    t.y = qp[k0 + 1] * scale;
    qf[s] = t;
  }

  v8f oacc[4];
#pragma unroll
  for (int i = 0; i < 4; ++i)
    oacc[i] = (v8f){0.f, 0.f, 0.f, 0.f, 0.f, 0.f, 0.f, 0.f};
  float m_i[8], l_i[8];
#pragma unroll
  for (int r = 0; r < 8; ++r) {
    m_i[r] = -__builtin_inff();
    l_i[r] = 0.f;
  }

  // async-stage one 64-key K/V block into buffer `buf`
  auto stage_kv = [&](int buf, int kv0) {
    const float* basek =
        qkv + (size_t)(b * TT + kv0) * (3 * CC) + CC + h * HDIM;
    const float* basev = basek + CC;
#pragma unroll
    for (int i = 0; i < 8; ++i) {
      int c = tid + i * 128;  // 0..1023 16B chunks
      int row = c >> 4, col = (c & 15) * 4;
      async_ld16(basek + (size_t)row * (3 * CC) + col,
                 &Ks[buf][row * LDK + col]);
      async_ld16(basev + (size_t)row * (3 * CC) + col,
                 &Vs[buf][row * LDV + col]);
    }
  };

  stage_kv(0, 0);
  async_wait0();
  __syncthreads();

  for (int kvb = 0; kvb <= qb; ++kvb) {  // causal: skip fully-masked blocks
    if (kvb < qb) stage_kv((kvb + 1) & 1, (kvb + 1) * 64);
    const float* Kb = &Ks[kvb & 1][0];
    const float* Vb = &Vs[kvb & 1][0];

    // S = (Q*scale) @ K^T  -> 16x64 strip per wave
    v8f sacc[4];
#pragma unroll
    for (int i = 0; i < 4; ++i)
      sacc[i] = (v8f){0.f, 0.f, 0.f, 0.f, 0.f, 0.f, 0.f, 0.f};
#pragma unroll
    for (int s = 0; s < 16; ++s) {
      int k0 = 4 * s + 2 * half;
      v2f af = qf[s];
#pragma unroll
      for (int sub = 0; sub < 4; ++sub) {
        v2f bf = *(const v2f*)&Kb[(16 * sub + ln) * LDK + k0];
        sacc[sub] = wmma4f32(af, bf, sacc[sub]);
      }
    }

    if (kvb == qb) {  // diagonal block: elementwise causal mask
#pragma unroll
      for (int sub = 0; sub < 4; ++sub)
#pragma unroll
        for (int r = 0; r < 8; ++r)
          if (16 * sub + ln > w * 16 + r + 8 * half)
            sacc[sub][r] = -__builtin_inff();
    }

    // online softmax: each row lives across the 16 lanes of a half
#pragma unroll
    for (int r = 0; r < 8; ++r) {
      float v = fmaxf(fmaxf(sacc[0][r], sacc[1][r]),
                      fmaxf(sacc[2][r], sacc[3][r]));
      v = fmaxf(v, __shfl_xor(v, 1));
      v = fmaxf(v, __shfl_xor(v, 2));
      v = fmaxf(v, __shfl_xor(v, 4));
      v = fmaxf(v, __shfl_xor(v, 8));
      float mn = fmaxf(m_i[r], v);
      float alpha = __expf(m_i[r] - mn);  // first block: exp(-inf)=0
      m_i[r] = mn;
      float rs = 0.f;
#pragma unroll
      for (int sub = 0; sub < 4; ++sub) {
        float p = __expf(sacc[sub][r] - mn);
        sacc[sub][r] = p;
        rs += p;
      }
      rs += __shfl_xor(rs, 1);
      rs += __shfl_xor(rs, 2);
      rs += __shfl_xor(rs, 4);
      rs += __shfl_xor(rs, 8);
      l_i[r] = l_i[r] * alpha + rs;
#pragma unroll
      for (int sub = 0; sub < 4; ++sub) oacc[sub][r] *= alpha;
    }

    // transpose P (D-layout) -> A-layout via wave-private LDS (in-order DS)
#pragma unroll
    for (int sub = 0; sub < 4; ++sub)
#pragma unroll
      for (int r = 0; r < 8; ++r)
        Ps[w][(r + 8 * half) * LDP + 16 * sub + ln] = sacc[sub][r];

    // O += P @ V
#pragma unroll
    for (int s = 0; s < 16; ++s) {
      int k0 = 4 * s + 2 * half;
      v2f af = *(const v2f*)&Ps[w][ln * LDP + k0];
#pragma unroll
      for (int sub = 0; sub < 4; ++sub) {
        v2f bf;
        bf.x = Vb[k0 * LDV + 16 * sub + ln];
        bf.y = Vb[(k0 + 1) * LDV + 16 * sub + ln];
        oacc[sub] = wmma4f32(af, bf, oacc[sub]);
      }
    }

    async_wait0();   // next K/V block resident
    __syncthreads(); // all waves done with current buffers
  }

  // normalize and write context [B,T,C] (heads at channel offset h*HD)
#pragma unroll
  for (int r = 0; r < 8; ++r) {
    float rinv = 1.0f / l_i[r];
    int row = q0 + w * 16 + r + 8 * half;
    float* o = ctx + (size_t)(b * TT + row) * CC + h * HDIM;
#pragma unroll
    for (int sub = 0; sub < 4; ++sub) o[16 * sub + ln] = oacc[sub][r] * rinv;
  }
}

// ---------------------------------------------------------------------------
extern "C" void kernel_launch(void* const* d_in, const int* in_sizes, int n_in,
                              void* d_out, int out_size, void* d_ws,
                              size_t ws_size, hipStream_t stream) {
  (void)in_sizes;
  (void)n_in;
  (void)out_size;
  (void)ws_size;
  const float* x = (const float*)d_in[0];
  // d_in[1] = attention_mask (tril) — causality applied analytically
  const float* Wqkv = (const float*)d_in[2];
  const float* bqkv = (const float*)d_in[3];
  const float* Wo = (const float*)d_in[4];
  const float* bo = (const float*)d_in[5];
  float* out = (float*)d_out;

  float* qkv = (float*)d_ws;               // [4096, 3072] = 50.3 MB
  float* ctx = qkv + (size_t)MM * 3 * CC;  // [4096, 1024] = 16.8 MB

  // 1) QKV projection: [4096,1024] @ [1024,3072] + bqkv
  gemm_bias_wmma<<<dim3((3 * CC) / 128, MM / 128), 256, 0, stream>>>(
      x, Wqkv, bqkv, qkv, MM, 3 * CC, CC);
  // 2) causal flash attention -> ctx [B,T,C]
  attn_wmma<<<dim3(TT / 64, HH, BB), 128, 0, stream>>>(qkv, ctx);
  // 3) output projection: [4096,1024] @ [1024,1024] + bo
  gemm_bias_wmma<<<dim3(CC / 128, MM / 128), 256, 0, stream>>>(ctx, Wo, bo, out,
                                                               MM, CC, CC);
}